// GraphProp_40072044871718
// MI455X (gfx1250) — compile-verified
//
#include <hip/hip_runtime.h>
#include <math.h>

typedef __attribute__((ext_vector_type(2))) float v2f;
typedef __attribute__((ext_vector_type(8))) float v8f;

#define NN 100000
#define EE 1600000
#define HH 64
#define NTILES (NN / 16)   // 6250, exact

static __device__ __forceinline__ v8f wmma4(v2f a, v2f b, v8f c) {
  // D = A(16x4,f32) x B(4x16,f32) + C(16x16,f32)
  return __builtin_amdgcn_wmma_f32_16x16x4_f32(
      /*neg_a=*/false, a, /*neg_b=*/false, b,
      /*c_mod=*/(short)0, c, /*reuse_a=*/false, /*reuse_b=*/false);
}

__global__ void zero_kernel(float* __restrict__ p, int n) {
  int i = blockIdx.x * blockDim.x + threadIdx.x;
  if (i < n) p[i] = 0.0f;
}

// One wave32 per edge: gather hv[src], scatter-add into s[dst]; lane0 does c,deg.
__global__ void edge_scatter_kernel(const float* __restrict__ hv,
                                    const float* __restrict__ he,
                                    const int* __restrict__ src,
                                    const int* __restrict__ dst,
                                    float* __restrict__ s,
                                    float* __restrict__ c,
                                    float* __restrict__ deg) {
  int gw = (blockIdx.x * blockDim.x + threadIdx.x) >> 5;
  int lane = threadIdx.x & 31;
  if (gw >= EE) return;
  int sv = src[gw];
  int dv = dst[gw];
  float h0 = hv[sv * HH + lane];
  float h1 = hv[sv * HH + 32 + lane];
  atomicAdd(&s[dv * HH + lane], h0);
  atomicAdd(&s[dv * HH + 32 + lane], h1);
  if (lane == 0) {
    atomicAdd(&c[dv], he[gw]);
    atomicAdd(&deg[dv], 1.0f);
  }
}

// One wave32 per 16-node tile.
// a = (deg .* hv) @ Wd^T + s @ Ws^T + deg (x) b_msg + c (x) w_e    [16 x 128]
// gi = a @ w_ih^T + b_ih ; gh = hv @ w_hh^T + b_hh ; GRUCell combine.
__global__ __launch_bounds__(128)
void node_update_kernel(const float* __restrict__ hv_in,
                        const float* __restrict__ s,
                        const float* __restrict__ c,
                        const float* __restrict__ deg,
                        const float* __restrict__ Wm,    // [128][129]
                        const float* __restrict__ bm,    // [128]
                        const float* __restrict__ wih,   // [192][128]
                        const float* __restrict__ whh,   // [192][64]
                        const float* __restrict__ bih,   // [192]
                        const float* __restrict__ bhh,   // [192]
                        float* __restrict__ hv_out) {
  __shared__ __align__(16) float a_lds[4][16 * 128];

  const int lane = threadIdx.x & 31;
  const int wv = threadIdx.x >> 5;
  int tile = blockIdx.x * 4 + wv;
  if (tile >= NTILES) tile = NTILES - 1;  // duplicate last tile: identical writes, no divergence
  const int m0 = tile * 16;

  const int nn  = lane & 15;        // A row / B,D col within tile
  const int kh2 = (lane >> 4) * 2;  // K sub-pair select for f32 fragments
  const int mA  = m0 + nn;          // global node row for A fragments
  const float degA = deg[mA];

  // ---------------- stage 1: acc = (deg.*h) @ Wd^T + s @ Ws^T ----------------
  v8f acc[8];
#pragma unroll
  for (int j = 0; j < 8; ++j) acc[j] = {};

  for (int k = 0; k < HH; k += 4) {
    const int ka = k + kh2;
    v2f ah = *reinterpret_cast<const v2f*>(&hv_in[mA * HH + ka]);
    ah.x *= degA;
    ah.y *= degA;
    const v2f as = *reinterpret_cast<const v2f*>(&s[mA * HH + ka]);
#pragma unroll
    for (int j = 0; j < 8; ++j) {
      const int o = j * 16 + nn;                // output column = row of Wd/Ws
      const float* wr = &Wm[o * 129 + ka];      // stride 129 (odd) -> scalar loads
      v2f bd; bd.x = wr[0];  bd.y = wr[1];      // Wd = W_msg[:, 0:64]
      v2f bs; bs.x = wr[64]; bs.y = wr[65];     // Ws = W_msg[:, 64:128]
      acc[j] = wmma4(ah, bd, acc[j]);
      acc[j] = wmma4(as, bs, acc[j]);
    }
  }

  // epilogue: add deg*b_msg + c*w_e, convert D-layout -> row-major LDS [16][128]
  const int r0 = kh2 << 2;  // D-layout row base: 0 or 8
  float degv[8], cv[8];
#pragma unroll
  for (int v = 0; v < 8; ++v) {
    degv[v] = deg[m0 + r0 + v];
    cv[v] = c[m0 + r0 + v];
  }
#pragma unroll
  for (int j = 0; j < 8; ++j) {
    const int o = j * 16 + nn;
    const float bmj = bm[o];
    const float wej = Wm[o * 129 + 128];        // w_e = W_msg[:, 128]
#pragma unroll
    for (int v = 0; v < 8; ++v) {
      a_lds[wv][(r0 + v) * 128 + o] = acc[j][v] + degv[v] * bmj + cv[v] * wej;
    }
  }
  __syncthreads();

  // ---------------- stage 2: GRU gates, one 16-column output tile at a time ----------------
  for (int j = 0; j < 4; ++j) {
    v8f ir = {}, iz = {}, in_ = {}, hr = {}, hz = {}, hn = {};

    // gi = a @ w_ih^T : K = 128
    for (int k = 0; k < 128; k += 4) {
      const int ka = k + kh2;
      const v2f aa = *reinterpret_cast<const v2f*>(&a_lds[wv][nn * 128 + ka]);
      const v2f br = *reinterpret_cast<const v2f*>(&wih[(j * 16 + nn) * 128 + ka]);
      const v2f bz = *reinterpret_cast<const v2f*>(&wih[(64 + j * 16 + nn) * 128 + ka]);
      const v2f bn = *reinterpret_cast<const v2f*>(&wih[(128 + j * 16 + nn) * 128 + ka]);
      ir  = wmma4(aa, br, ir);
      iz  = wmma4(aa, bz, iz);
      in_ = wmma4(aa, bn, in_);
    }
    // gh = h @ w_hh^T : K = 64
    for (int k = 0; k < HH; k += 4) {
      const int ka = k + kh2;
      const v2f ha = *reinterpret_cast<const v2f*>(&hv_in[mA * HH + ka]);
      const v2f br = *reinterpret_cast<const v2f*>(&whh[(j * 16 + nn) * HH + ka]);
      const v2f bz = *reinterpret_cast<const v2f*>(&whh[(64 + j * 16 + nn) * HH + ka]);
      const v2f bn = *reinterpret_cast<const v2f*>(&whh[(128 + j * 16 + nn) * HH + ka]);
      hr = wmma4(ha, br, hr);
      hz = wmma4(ha, bz, hz);
      hn = wmma4(ha, bn, hn);
    }

    // exact GRUCell elementwise (fp32)
    const int oc = j * 16 + nn;  // output column in H
    const float bir = bih[oc], biz = bih[64 + oc], bin = bih[128 + oc];
    const float bhr = bhh[oc], bhz = bhh[64 + oc], bhn = bhh[128 + oc];
#pragma unroll
    for (int v = 0; v < 8; ++v) {
      const int gm = m0 + r0 + v;  // D-layout row -> global node
      const float r  = 1.0f / (1.0f + __expf(-(ir[v] + bir + hr[v] + bhr)));
      const float z  = 1.0f / (1.0f + __expf(-(iz[v] + biz + hz[v] + bhz)));
      const float nt = tanhf(in_[v] + bin + r * (hn[v] + bhn));
      const float hold = hv_in[gm * HH + oc];
      hv_out[gm * HH + oc] = (1.0f - z) * nt + z * hold;
    }
  }
}

extern "C" void kernel_launch(void* const* d_in, const int* in_sizes, int n_in,
                              void* d_out, int out_size, void* d_ws, size_t ws_size,
                              hipStream_t stream) {
  const float* hv    = (const float*)d_in[0];
  const float* he    = (const float*)d_in[1];
  const int*   src   = (const int*)d_in[2];
  const int*   dst   = (const int*)d_in[3];
  const float* W_msg = (const float*)d_in[4];  // [T][128][129]
  const float* b_msg = (const float*)d_in[5];  // [T][128]
  const float* w_ih  = (const float*)d_in[6];  // [T][192][128]
  const float* w_hh  = (const float*)d_in[7];  // [T][192][64]
  const float* b_ih  = (const float*)d_in[8];  // [T][192]
  const float* b_hh  = (const float*)d_in[9];  // [T][192]

  float* ws     = (float*)d_ws;
  float* s      = ws;                       // [N][64]
  float* c      = s + (size_t)NN * HH;      // [N]
  float* deg    = c + NN;                   // [N]
  float* hv_tmp = deg + NN;                 // [N][64]
  float* out    = (float*)d_out;

  const int zn = NN * (HH + 2);             // s + c + deg contiguous
  for (int t = 0; t < 2; ++t) {
    const float* hin = (t == 0) ? hv : hv_tmp;
    float* hout      = (t == 0) ? hv_tmp : out;

    zero_kernel<<<(zn + 255) / 256, 256, 0, stream>>>(s, zn);
    edge_scatter_kernel<<<(EE * 32) / 256, 256, 0, stream>>>(hin, he, src, dst, s, c, deg);
    node_update_kernel<<<(NTILES + 3) / 4, 128, 0, stream>>>(
        hin, s, c, deg,
        W_msg + (size_t)t * 128 * 129, b_msg + (size_t)t * 128,
        w_ih + (size_t)t * 192 * 128, w_hh + (size_t)t * 192 * 64,
        b_ih + (size_t)t * 192, b_hh + (size_t)t * 192, hout);
  }
}